// SamplerBase_56564719288551
// MI455X (gfx1250) — compile-verified
//
#include <hip/hip_runtime.h>

// ---------------------------------------------------------------------------
// Sampler: top-p(0.9) + top-k(50) filter, neg-entropy confidence, categorical
// pick, on [1024 x 128000] fp32 logits.
//
// Per row (one workgroup, 256 threads = 8 wave32):
//   Phase 1: stream the row through a double-buffered LDS stage fed by the
//            CDNA5 Tensor Data Mover (tensor_load_to_lds / TENSORcnt /
//            s_wait_tensorcnt). Row max + 4096-bin histogram of the monotonic
//            float-bit key (ds_add atomics) => radix-select top-50 threshold.
//   Phase 2: re-stream (row is 512 KB; hot in the 192 MB L2). Full
//            sum(exp(z-m)) + collect candidates with key >= threshold.
//   Phase 3: bitonic sort <=256 candidates, shift-by-one top-p rule against
//            the FULL softmax mass, cut to k=50, entropy + categorical pick.
//
// HBM bound: <= 1 GB traffic => ~44 us at 23.3 TB/s. LDS consume side uses
// ds_load_b128 (float4) so the 8 consumer waves outrun the TDM easily.
// ---------------------------------------------------------------------------

#define N_ROWS   1024
#define BLOCK    256          // 8 waves of 32
#define TILE     4000         // floats per TDM tile (16 KB); 128000 = 32 tiles
#define NBINS    4096         // histogram over top 12 bits of the float key
#define TOPK     50
#define TOPP     0.9f
#define EPSV     1e-10f
#define CANDMAX  256          // candidate cap == BLOCK (one slot per thread)
#define FLOATMIN (-3.402823466e38f)

typedef __attribute__((ext_vector_type(4))) unsigned int u32x4;
typedef __attribute__((ext_vector_type(4))) int          i32x4;
typedef __attribute__((ext_vector_type(8))) int          i32x8;

#if __has_builtin(__builtin_amdgcn_tensor_load_to_lds) && \
    __has_builtin(__builtin_amdgcn_s_wait_tensorcnt)
#define HAVE_TDM 1
#else
#define HAVE_TDM 0
#endif

// Monotonic map: float -> uint32 preserving order.
__device__ __forceinline__ unsigned fkey(float f) {
  unsigned b = __float_as_uint(f);
  return (b & 0x80000000u) ? ~b : (b | 0x80000000u);
}

// Low 32 bits of a flat pointer into LDS == LDS byte offset.
__device__ __forceinline__ unsigned lds_byte_off(const void* p) {
  return (unsigned)(unsigned long long)(uintptr_t)p;
}

__device__ __forceinline__ unsigned mix_hash(unsigned s) {
  s ^= s >> 16; s *= 0x85EBCA6Bu;
  s ^= s >> 13; s *= 0xC2B2AE35u;
  s ^= s >> 16; return s;
}

#if HAVE_TDM
// Issue one TDM load of `nelem` fp32 global -> LDS (D# per CDNA5 ISA sec. 8).
__device__ __forceinline__ void tdm_load_tile(const float* gsrc,
                                              unsigned lds_off, int nelem) {
  unsigned long long ga = (unsigned long long)(uintptr_t)gsrc;
  u32x4 g0;
  g0[0] = 1u;                                             // count=1, user D#
  g0[1] = lds_off;                                        // lds_addr (bytes)
  g0[2] = (unsigned)(ga & 0xFFFFFFFFu);                   // global_addr[31:0]
  g0[3] = (unsigned)((ga >> 32) & 0x01FFFFFFu)            // global_addr[56:32]
        | (2u << 30);                                     // type = 2 ("image")
  unsigned ne = (unsigned)nelem;
  i32x8 g1;
  g1[0] = (int)(2u << 16);                                // data_size=4B, mask=0
  g1[1] = (int)((ne & 0xFFFFu) << 16);                    // tensor_dim0[15:0]
  g1[2] = (int)(((ne >> 16) & 0xFFFFu) | (1u << 16));     // dim0[31:16], dim1=1
  g1[3] = (int)((ne & 0xFFFFu) << 16);                    // tile_dim0 = nelem
  g1[4] = 1;                                              // tile_dim1=1, dim2=0
  g1[5] = (int)ne;                                        // dim0_stride[31:0]
  g1[6] = 0;
  g1[7] = 0;
  i32x4 z4 = {0, 0, 0, 0};
#if defined(__clang_major__) && (__clang_major__ >= 23)
  i32x8 z8 = {0, 0, 0, 0, 0, 0, 0, 0};
  __builtin_amdgcn_tensor_load_to_lds(g0, g1, z4, z4, z8, 0);
#else
  __builtin_amdgcn_tensor_load_to_lds(g0, g1, z4, z4, 0);
#endif
}
#endif

__global__ __launch_bounds__(BLOCK)
void sampler_topk_topp_entropy(const float* __restrict__ logits,
                               float* __restrict__ out, int V) {
  __shared__ alignas(16) float buf[2][TILE];
  __shared__ unsigned hist[NBINS];
  __shared__ float    red[BLOCK];
  __shared__ float    cand_v[CANDMAX];
  __shared__ int      cand_i[CANDMAX];
  __shared__ float    cand_e[CANDMAX];
  __shared__ unsigned s_cnt;
  __shared__ unsigned s_thr;
  __shared__ float    s_max;
  __shared__ float    s_sum;

  const int tid = threadIdx.x;
  const int row = blockIdx.x;
  const float* __restrict__ rowp = logits + (long long)row * V;
  const int nt = (V + TILE - 1) / TILE;

  for (int i = tid; i < NBINS; i += BLOCK) hist[i] = 0u;
  if (tid == 0) s_cnt = 0u;
  __syncthreads();

  // ------------------- Phase 1: row max + key histogram -------------------
  float lmax = FLOATMIN;
#if HAVE_TDM
  if (tid < 32) tdm_load_tile(rowp, lds_byte_off(&buf[0][0]),
                              (V < TILE) ? V : TILE);
  for (int t = 0; t < nt; ++t) {
    const int p  = t & 1;
    const int ne = (V - t * TILE < TILE) ? (V - t * TILE) : TILE;
    if (tid < 32) {                       // wave 0 drives the DMA pipeline
      if (t + 1 < nt) {
        const int ne2 = (V - (t + 1) * TILE < TILE) ? (V - (t + 1) * TILE) : TILE;
        tdm_load_tile(rowp + (long long)(t + 1) * TILE,
                      lds_byte_off(&buf[(t + 1) & 1][0]), ne2);
        __builtin_amdgcn_s_wait_tensorcnt(1);   // tile t complete (in-order)
      } else {
        __builtin_amdgcn_s_wait_tensorcnt(0);   // drain
      }
    }
    __syncthreads();                       // publish tile t to all waves
    const int nev = ne & ~3;               // vector body (ds_load_b128)
    for (int i = tid * 4; i < nev; i += BLOCK * 4) {
      float4 v = *(const float4*)&buf[p][i];
      atomicAdd(&hist[fkey(v.x) >> 20], 1u);
      atomicAdd(&hist[fkey(v.y) >> 20], 1u);
      atomicAdd(&hist[fkey(v.z) >> 20], 1u);
      atomicAdd(&hist[fkey(v.w) >> 20], 1u);
      lmax = fmaxf(lmax, fmaxf(fmaxf(v.x, v.y), fmaxf(v.z, v.w)));
    }
    for (int i = nev + tid; i < ne; i += BLOCK) {   // tail
      float f = buf[p][i];
      atomicAdd(&hist[fkey(f) >> 20], 1u);
      lmax = fmaxf(lmax, f);
    }
    __syncthreads();                       // buffer p reusable by DMA
  }
#else
  for (int t = 0; t < nt; ++t) {
    const int ne = (V - t * TILE < TILE) ? (V - t * TILE) : TILE;
    for (int i = tid; i < ne; i += BLOCK) buf[0][i] = rowp[(long long)t * TILE + i];
    __syncthreads();
    for (int i = tid; i < ne; i += BLOCK) {
      float f = buf[0][i];
      atomicAdd(&hist[fkey(f) >> 20], 1u);
      lmax = fmaxf(lmax, f);
    }
    __syncthreads();
  }
#endif

  red[tid] = lmax;
  __syncthreads();
  for (int s = BLOCK / 2; s > 0; s >>= 1) {
    if (tid < s) red[tid] = fmaxf(red[tid], red[tid + s]);
    __syncthreads();
  }
  if (tid == 0) {
    s_max = red[0];
    unsigned acc = 0; int b = 0;
    for (int i = NBINS - 1; i >= 0; --i) {       // suffix scan: k-th key bin
      acc += hist[i];
      if (acc >= TOPK) { b = i; break; }
    }
    s_thr = (unsigned)b << 20;
  }
  __syncthreads();
  const float    m   = s_max;
  const unsigned thr = s_thr;

  // ------------- Phase 2: full sum(exp) + candidate collection ------------
  float lsum = 0.f;
#if HAVE_TDM
  if (tid < 32) tdm_load_tile(rowp, lds_byte_off(&buf[0][0]),
                              (V < TILE) ? V : TILE);
  for (int t = 0; t < nt; ++t) {
    const int p  = t & 1;
    const int ne = (V - t * TILE < TILE) ? (V - t * TILE) : TILE;
    if (tid < 32) {
      if (t + 1 < nt) {
        const int ne2 = (V - (t + 1) * TILE < TILE) ? (V - (t + 1) * TILE) : TILE;
        tdm_load_tile(rowp + (long long)(t + 1) * TILE,
                      lds_byte_off(&buf[(t + 1) & 1][0]), ne2);
        __builtin_amdgcn_s_wait_tensorcnt(1);
      } else {
        __builtin_amdgcn_s_wait_tensorcnt(0);
      }
    }
    __syncthreads();
    const int base = t * TILE;
    const int nev  = ne & ~3;
    for (int i = tid * 4; i < nev; i += BLOCK * 4) {
      float4 v = *(const float4*)&buf[p][i];
      lsum += __expf(v.x - m) + __expf(v.y - m) +
              __expf(v.z - m) + __expf(v.w - m);
      #pragma unroll
      for (int j = 0; j < 4; ++j) {
        float f = (j == 0) ? v.x : (j == 1) ? v.y : (j == 2) ? v.z : v.w;
        if (fkey(f) >= thr) {
          unsigned pos = atomicAdd(&s_cnt, 1u);
          if (pos < CANDMAX) { cand_v[pos] = f; cand_i[pos] = base + i + j; }
        }
      }
    }
    for (int i = nev + tid; i < ne; i += BLOCK) {   // tail
      float f = buf[p][i];
      lsum += __expf(f - m);
      if (fkey(f) >= thr) {
        unsigned pos = atomicAdd(&s_cnt, 1u);
        if (pos < CANDMAX) { cand_v[pos] = f; cand_i[pos] = base + i; }
      }
    }
    __syncthreads();
  }
#else
  for (int t = 0; t < nt; ++t) {
    const int ne = (V - t * TILE < TILE) ? (V - t * TILE) : TILE;
    for (int i = tid; i < ne; i += BLOCK) buf[0][i] = rowp[(long long)t * TILE + i];
    __syncthreads();
    for (int i = tid; i < ne; i += BLOCK) {
      float f = buf[0][i];
      lsum += __expf(f - m);
      if (fkey(f) >= thr) {
        unsigned pos = atomicAdd(&s_cnt, 1u);
        if (pos < CANDMAX) { cand_v[pos] = f; cand_i[pos] = t * TILE + i; }
      }
    }
    __syncthreads();
  }
#endif

  red[tid] = lsum;
  __syncthreads();
  for (int s = BLOCK / 2; s > 0; s >>= 1) {
    if (tid < s) red[tid] += red[tid + s];
    __syncthreads();
  }
  if (tid == 0) s_sum = red[0];
  __syncthreads();

  // ------------------- Phase 3: sort + filter + epilogue ------------------
  unsigned nc = s_cnt; if (nc > CANDMAX) nc = CANDMAX;
  if ((unsigned)tid >= nc) { cand_v[tid] = FLOATMIN; cand_i[tid] = 0; }
  __syncthreads();

  // Bitonic sort CANDMAX elements, descending by value (index rides along).
  for (int k = 2; k <= CANDMAX; k <<= 1) {
    for (int j = k >> 1; j > 0; j >>= 1) {
      int ixj = tid ^ j;
      if (ixj > tid) {
        float a = cand_v[tid], b = cand_v[ixj];
        bool sw = ((tid & k) == 0) ? (a < b) : (a > b);
        if (sw) {
          cand_v[tid] = b; cand_v[ixj] = a;
          int ti = cand_i[tid]; cand_i[tid] = cand_i[ixj]; cand_i[ixj] = ti;
        }
      }
      __syncthreads();
    }
  }

  cand_e[tid] = __expf(cand_v[tid] - m);     // exp(FLOATMIN - m) == 0
  __syncthreads();

  if (tid == 0) {
    const float Sfull = s_sum;
    const int   n     = (int)nc;
    // top-p shift-by-one keep rule against the FULL softmax mass
    float cum = 0.f; int L = 0;
    for (int i = 0; i < n; ++i) {
      if (i > 0 && cum > TOPP) break;
      cum += cand_e[i] / Sfull;
      ++L;
    }
    if (L > TOPK) L = TOPK;
    if (L < 1)    L = 1;
    float SK = 0.f;
    for (int i = 0; i < L; ++i) SK += cand_e[i];
    const float inv = 1.f / SK;
    float conf = 0.f;
    for (int i = 0; i < L; ++i) {
      float pp = cand_e[i] * inv;
      conf += pp * __logf(pp + EPSV);
    }
    // deterministic categorical draw over the kept prefix
    unsigned h   = mix_hash((unsigned)row * 0x9E3779B9u + 42u);
    float    u01 = (float)(h >> 8) * (1.0f / 16777216.0f);
    float targ = u01 * SK, c = 0.f; int j = L - 1;
    for (int i = 0; i < L; ++i) { c += cand_e[i]; if (c >= targ) { j = i; break; } }
    out[row]              = conf;                   // confidence
    out[N_ROWS + row]     = (float)cand_i[j];       // x0
    out[2 * N_ROWS + row] = cand_e[j] * inv;        // initial_confidence
  }
}

extern "C" void kernel_launch(void* const* d_in, const int* in_sizes, int n_in,
                              void* d_out, int out_size, void* d_ws, size_t ws_size,
                              hipStream_t stream) {
  (void)n_in; (void)out_size; (void)d_ws; (void)ws_size;
  const float* logits = (const float*)d_in[0];
  float* out = (float*)d_out;
  const int V = in_sizes[0] / N_ROWS;
  sampler_topk_topp_entropy<<<N_ROWS, BLOCK, 0, stream>>>(logits, out, V);
}